// InceptionBlock_28999619182972
// MI455X (gfx1250) — compile-verified
//
#include <hip/hip_runtime.h>
#include <math.h>

typedef float v2f __attribute__((ext_vector_type(2)));
typedef float v8f __attribute__((ext_vector_type(8)));

#define LRELU_GAT 0.2f
#define LRELU_OUT 0.01f
#define BN_EPS    1e-5f

static __host__ __device__ inline long cdivl(long a, long b) { return (a + b - 1) / b; }

// ---------------- fill kernels ----------------
__global__ void k_fill_f32(float* __restrict__ p, float v, long n) {
  long i = (long)blockIdx.x * blockDim.x + threadIdx.x;
  if (i < n) p[i] = v;
}
__global__ void k_fill_u32(unsigned* __restrict__ p, unsigned v, long n) {
  long i = (long)blockIdx.x * blockDim.x + threadIdx.x;
  if (i < n) p[i] = v;
}

// ---------------- graph normalization ----------------
__global__ void k_degree(const int* __restrict__ src, float* __restrict__ deg, int E) {
  int e = blockIdx.x * blockDim.x + threadIdx.x;
  if (e < E) atomicAdd(&deg[src[e]], 1.0f);
}
__global__ void k_dis(const float* __restrict__ deg, float* __restrict__ dis, int N) {
  int i = blockIdx.x * blockDim.x + threadIdx.x;
  if (i < N) {
    float d = deg[i];
    dis[i] = d > 0.f ? rsqrtf(fmaxf(d, 1.f)) : 0.f;
  }
}
__global__ void k_wedge(const int* __restrict__ src, const int* __restrict__ dst,
                        const float* __restrict__ dis, float* __restrict__ w, int E) {
  int e = blockIdx.x * blockDim.x + threadIdx.x;
  if (e < E) w[e] = -dis[src[e]] * dis[dst[e]];
}

// ---------------- Cheb propagation: P[dst] += T[src] * w ----------------
// 8 threads per edge, float4 per thread (32 channels)
__global__ void k_prop(const float* __restrict__ T, const int* __restrict__ src,
                       const int* __restrict__ dst, const float* __restrict__ w,
                       float* __restrict__ P, int E) {
  long idx = (long)blockIdx.x * blockDim.x + threadIdx.x;
  int e = (int)(idx >> 3);
  if (e >= E) return;
  int cg = ((int)idx & 7) * 4;
  float we = w[e];
  const float4 v = *(const float4*)(T + src[e] * 32 + cg);
  float* p = P + dst[e] * 32 + cg;
  atomicAdd(p + 0, v.x * we);
  atomicAdd(p + 1, v.y * we);
  atomicAdd(p + 2, v.z * we);
  atomicAdd(p + 3, v.w * we);
}
// Tx2 = 2*P - Tx0
__global__ void k_combine(float* __restrict__ T2, const float* __restrict__ P,
                          const float* __restrict__ T0, long n) {
  long i = (long)blockIdx.x * blockDim.x + threadIdx.x;
  if (i < n) T2[i] = 2.f * P[i] - T0[i];
}
// Out[n][c] = b[c]
__global__ void k_bias_init(float* __restrict__ Out, const float* __restrict__ b, long n32) {
  long i = (long)blockIdx.x * blockDim.x + threadIdx.x;
  if (i < n32) Out[i] = b[i & 31];
}

// ---------------- FP32 WMMA GEMM: Out[N,NCOL] (+)= A[N,32] @ W[32,NCOL] ----------------
// One wave per 16x32 output tile; V_WMMA_F32_16X16X4_F32, 8 k-steps cover K=32.
// All indexing is 32-bit (max offset ~6.4M elements) to avoid 64-bit VALU address math.
__global__ void k_gemm_wmma(const float* __restrict__ A, const float* __restrict__ W,
                            float* __restrict__ Out, int N, int NCOL, int accum) {
  const int lane = threadIdx.x & 31;
  const int wave = threadIdx.x >> 5;
  const int row0 = (blockIdx.x * 4 + wave) * 16;
  if (row0 >= N) return;                       // uniform per wave; N%16==0 so no partial tiles
  const int colBase = blockIdx.y * 32;
  const int m  = lane & 15;                    // A-row within tile
  const int kh = (lane >> 4) * 2;              // K pair: lanes 0-15 -> K0/K1, lanes 16-31 -> K2/K3
  const int cn = lane & 15;                    // column within 16-col block
  const int mr = (lane >> 4) * 8;              // C-row offset for high half-wave

  const float* __restrict__ Arow = A + (row0 + m) * 32 + kh;   // 8B-aligned pair base
  const float* __restrict__ Wp   = W + kh * NCOL + colBase + cn;
  float* __restrict__ OutP       = Out + (row0 + mr) * NCOL + colBase + cn;

  v8f c0 = {0.f,0.f,0.f,0.f,0.f,0.f,0.f,0.f};
  v8f c1 = {0.f,0.f,0.f,0.f,0.f,0.f,0.f,0.f};
  if (accum) {
#pragma unroll
    for (int v = 0; v < 8; ++v) {
      c0[v] = OutP[v * NCOL];
      c1[v] = OutP[v * NCOL + 16];
    }
  }
#pragma unroll
  for (int kk = 0; kk < 8; ++kk) {
    v2f a = *(const v2f*)(Arow + kk * 4);      // contiguous K pair -> global_load_b64
    const float* wq = Wp + kk * 4 * NCOL;
    v2f b0, b1;
    b0[0] = wq[0];
    b0[1] = wq[NCOL];
    b1[0] = wq[16];
    b1[1] = wq[NCOL + 16];
    c0 = __builtin_amdgcn_wmma_f32_16x16x4_f32(false, a, false, b0, (short)0, c0, false, false);
    c1 = __builtin_amdgcn_wmma_f32_16x16x4_f32(false, a, false, b1, (short)0, c1, false, false);
  }
#pragma unroll
  for (int v = 0; v < 8; ++v) {
    OutP[v * NCOL]      = c0[v];
    OutP[v * NCOL + 16] = c1[v];
  }
}

// ---------------- SuperGAT edge passes ----------------
__device__ inline unsigned f32_key(float f) {      // monotone float -> uint mapping
  unsigned b = __float_as_uint(f);
  return (b & 0x80000000u) ? ~b : (b | 0x80000000u);
}
__device__ inline float key_f32(unsigned k) {
  return (k & 0x80000000u) ? __uint_as_float(k & 0x7FFFFFFFu) : __uint_as_float(~k);
}

// pass 1: alpha[e,h] = leaky(( <xj,att_l> + <xi,att_r> ) * sigmoid(<xi,xj>)); segment max via key
__global__ void k_gat_alpha(const float* __restrict__ H, const int* __restrict__ src,
                            const int* __restrict__ dst, const float* __restrict__ att_l,
                            const float* __restrict__ att_r, float* __restrict__ alpha,
                            unsigned* __restrict__ mkey, int E, int E2) {
  long idx = (long)blockIdx.x * blockDim.x + threadIdx.x;
  int e = (int)(idx >> 1);
  if (e >= E2) return;
  int hd = (int)idx & 1;
  int s = (e < E) ? src[e] : (e - E);
  int d = (e < E) ? dst[e] : (e - E);
  const float4* xj = (const float4*)(H + s * 64 + hd * 32);
  const float4* xi = (const float4*)(H + d * 64 + hd * 32);
  const float4* aL = (const float4*)(att_l + hd * 32);
  const float4* aR = (const float4*)(att_r + hd * 32);
  float logit = 0.f, al = 0.f, ar = 0.f;
#pragma unroll
  for (int q = 0; q < 8; ++q) {
    float4 j = xj[q], i = xi[q], L = aL[q], R = aR[q];
    logit += i.x * j.x + i.y * j.y + i.z * j.z + i.w * j.w;
    al    += j.x * L.x + j.y * L.y + j.z * L.z + j.w * L.w;
    ar    += i.x * R.x + i.y * R.y + i.z * R.z + i.w * R.w;
  }
  float a = (al + ar) * (1.f / (1.f + __expf(-logit)));
  a = a > 0.f ? a : LRELU_GAT * a;
  alpha[e * 2 + hd] = a;
  atomicMax(&mkey[d * 2 + hd], f32_key(a));
}

// pass 2: e = exp(alpha - m[dst]); segment sum
__global__ void k_gat_exp(const int* __restrict__ src, const int* __restrict__ dst,
                          float* __restrict__ alpha, const unsigned* __restrict__ mkey,
                          float* __restrict__ ssum, int E, int E2) {
  long idx = (long)blockIdx.x * blockDim.x + threadIdx.x;
  int e = (int)(idx >> 1);
  if (e >= E2) return;
  int hd = (int)idx & 1;
  int d = (e < E) ? dst[e] : (e - E);
  float m = key_f32(mkey[d * 2 + hd]);
  float ex = __expf(alpha[e * 2 + hd] - m);
  alpha[e * 2 + hd] = ex;
  atomicAdd(&ssum[d * 2 + hd], ex);
}

// pass 3: G[dst,h,:] += xj * (e / s[dst]); 16 threads per edge (2 heads x 8 float4 groups)
__global__ void k_gat_scatter(const float* __restrict__ H, const int* __restrict__ src,
                              const int* __restrict__ dst, const float* __restrict__ alpha,
                              const float* __restrict__ ssum, float* __restrict__ G,
                              int E, int E2) {
  long idx = (long)blockIdx.x * blockDim.x + threadIdx.x;
  int e = (int)(idx >> 4);
  if (e >= E2) return;
  int rem = (int)idx & 15;
  int hd = rem >> 3;
  int cg = (rem & 7) * 4;
  int s = (e < E) ? src[e] : (e - E);
  int d = (e < E) ? dst[e] : (e - E);
  float a = alpha[e * 2 + hd] / ssum[d * 2 + hd];
  const float4 v = *(const float4*)(H + s * 64 + hd * 32 + cg);
  float* g = G + d * 64 + hd * 32 + cg;
  atomicAdd(g + 0, v.x * a);
  atomicAdd(g + 1, v.y * a);
  atomicAdd(g + 2, v.z * a);
  atomicAdd(g + 3, v.w * a);
}

// ---------------- post: head mean + bias, leaky+residual, BN stats (LDS-reduced) ----------------
// 256 threads = 8 nodes x 32 channels per block
__global__ void k_post_stats(const float* __restrict__ G, const float* __restrict__ gat_b,
                             float* __restrict__ Y, float* __restrict__ bns,
                             float* __restrict__ bns2, int N) {
  __shared__ float s1[32], s2[32];
  int c = threadIdx.x & 31;
  int local = threadIdx.x >> 5;  // 0..7 nodes per block
  if (threadIdx.x < 32) { s1[c] = 0.f; s2[c] = 0.f; }
  __syncthreads();
  int n = blockIdx.x * 8 + local;
  float y = 0.f;
  if (n < N) {
    float x2 = 0.5f * (G[n * 64 + c] + G[n * 64 + 32 + c]) + gat_b[c];
    float l = x2 > 0.f ? x2 : LRELU_OUT * x2;
    y = l + x2;
    Y[n * 32 + c] = y;
  }
  atomicAdd(&s1[c], y);
  atomicAdd(&s2[c], y * y);
  __syncthreads();
  if (threadIdx.x < 32) { atomicAdd(&bns[c], s1[c]); atomicAdd(&bns2[c], s2[c]); }
}

__global__ void k_bn_norm(const float* __restrict__ Y, const float* __restrict__ bns,
                          const float* __restrict__ bns2, const float* __restrict__ g,
                          const float* __restrict__ b, float* __restrict__ out, int N) {
  long i = (long)blockIdx.x * blockDim.x + threadIdx.x;
  if (i >= (long)N * 32) return;
  int c = (int)(i & 31);
  float invN = 1.f / (float)N;
  float mu = bns[c] * invN;
  float var = bns2[c] * invN - mu * mu;
  out[i] = (Y[i] - mu) * rsqrtf(var + BN_EPS) * g[c] + b[c];
}

// ---------------- launcher ----------------
extern "C" void kernel_launch(void* const* d_in, const int* in_sizes, int n_in,
                              void* d_out, int out_size, void* d_ws, size_t ws_size,
                              hipStream_t stream) {
  const float* x_in   = (const float*)d_in[0];
  const int*   eidx   = (const int*)d_in[1];
  // d_in[2] = edge_attr: unused by the reference
  const float* cheb_w = (const float*)d_in[3];
  const float* cheb_b = (const float*)d_in[4];
  const float* lin_w  = (const float*)d_in[5];
  const float* att_l  = (const float*)d_in[6];
  const float* att_r  = (const float*)d_in[7];
  const float* gat_b  = (const float*)d_in[8];
  const float* bn_g   = (const float*)d_in[9];
  const float* bn_b   = (const float*)d_in[10];

  const long N  = in_sizes[0] / 32;
  const long E  = in_sizes[1] / 2;
  const long E2 = E + N;
  const int* src = eidx;
  const int* dst = eidx + E;

  float* ws = (float*)d_ws;
  float* DEG = ws;  ws += N;
  float* DIS = ws;  ws += N;
  float* WED = ws;  ws += E;
  float* NB0 = ws;  ws += N * 32;
  float* NB1 = ws;  ws += N * 32;   // propagation scratch P
  float* NB2 = ws;  ws += N * 32;
  float* NB3 = ws;  ws += N * 32;
  float* X   = ws;  ws += N * 32;   // ChebConv output
  float* H   = ws;  ws += N * 64;   // lin projection [N, 2, 32]
  float* ALP = ws;  ws += E2 * 2;   // alpha / exp
  unsigned* MKEY = (unsigned*)ws; ws += N * 2;
  float* SSM = ws;  ws += N * 2;
  float* G   = ws;  ws += N * 64;   // GAT accumulation
  float* Y   = ws;  ws += N * 32;
  float* BNS = ws;  ws += 32;
  float* BN2 = ws;  ws += 32;

  const int TB = 256;
  const long n32 = N * 32;
  const int rowTiles = (int)cdivl(N, 16);
  dim3 gemmGrid32((unsigned)cdivl(rowTiles, 4), 1);
  dim3 gemmGrid64((unsigned)cdivl(rowTiles, 4), 2);

#define GRID(n) dim3((unsigned)cdivl((long)(n), TB))

  // --- graph normalization weights ---
  k_fill_f32<<<GRID(N), TB, 0, stream>>>(DEG, 0.f, N);
  k_degree<<<GRID(E), TB, 0, stream>>>(src, DEG, (int)E);
  k_dis<<<GRID(N), TB, 0, stream>>>(DEG, DIS, (int)N);
  k_wedge<<<GRID(E), TB, 0, stream>>>(src, dst, DIS, WED, (int)E);

  // --- ChebConv: X = sum_k Tk(x) @ W_k + b ---
  k_bias_init<<<GRID(n32), TB, 0, stream>>>(X, cheb_b, n32);
  // k=0: Tx0 = x
  k_gemm_wmma<<<gemmGrid32, 128, 0, stream>>>(x_in, cheb_w + 0 * 1024, X, (int)N, 32, 1);
  // k=1: Tx1 = prop(x) -> NB0
  k_fill_f32<<<GRID(n32), TB, 0, stream>>>(NB0, 0.f, n32);
  k_prop<<<GRID(E * 8), TB, 0, stream>>>(x_in, src, dst, WED, NB0, (int)E);
  k_gemm_wmma<<<gemmGrid32, 128, 0, stream>>>(NB0, cheb_w + 1 * 1024, X, (int)N, 32, 1);
  // k=2: Tx2 = 2*prop(NB0) - x -> NB2
  k_fill_f32<<<GRID(n32), TB, 0, stream>>>(NB1, 0.f, n32);
  k_prop<<<GRID(E * 8), TB, 0, stream>>>(NB0, src, dst, WED, NB1, (int)E);
  k_combine<<<GRID(n32), TB, 0, stream>>>(NB2, NB1, x_in, n32);
  k_gemm_wmma<<<gemmGrid32, 128, 0, stream>>>(NB2, cheb_w + 2 * 1024, X, (int)N, 32, 1);
  // k=3: Tx3 = 2*prop(NB2) - NB0 -> NB3
  k_fill_f32<<<GRID(n32), TB, 0, stream>>>(NB1, 0.f, n32);
  k_prop<<<GRID(E * 8), TB, 0, stream>>>(NB2, src, dst, WED, NB1, (int)E);
  k_combine<<<GRID(n32), TB, 0, stream>>>(NB3, NB1, NB0, n32);
  k_gemm_wmma<<<gemmGrid32, 128, 0, stream>>>(NB3, cheb_w + 3 * 1024, X, (int)N, 32, 1);
  // k=4: Tx4 = 2*prop(NB3) - NB2 -> NB0 (reuse)
  k_fill_f32<<<GRID(n32), TB, 0, stream>>>(NB1, 0.f, n32);
  k_prop<<<GRID(E * 8), TB, 0, stream>>>(NB3, src, dst, WED, NB1, (int)E);
  k_combine<<<GRID(n32), TB, 0, stream>>>(NB0, NB1, NB2, n32);
  k_gemm_wmma<<<gemmGrid32, 128, 0, stream>>>(NB0, cheb_w + 4 * 1024, X, (int)N, 32, 1);

  // --- SuperGAT ---
  k_gemm_wmma<<<gemmGrid64, 128, 0, stream>>>(X, lin_w, H, (int)N, 64, 0);
  k_fill_u32<<<GRID(N * 2), TB, 0, stream>>>(MKEY, 0u, N * 2);
  k_gat_alpha<<<GRID(E2 * 2), TB, 0, stream>>>(H, src, dst, att_l, att_r, ALP, MKEY, (int)E, (int)E2);
  k_fill_f32<<<GRID(N * 2), TB, 0, stream>>>(SSM, 0.f, N * 2);
  k_gat_exp<<<GRID(E2 * 2), TB, 0, stream>>>(src, dst, ALP, MKEY, SSM, (int)E, (int)E2);
  k_fill_f32<<<GRID(N * 64), TB, 0, stream>>>(G, 0.f, N * 64);
  k_gat_scatter<<<GRID(E2 * 16), TB, 0, stream>>>(H, src, dst, ALP, SSM, G, (int)E, (int)E2);

  // --- head mean + bias + leaky + residual + batchnorm ---
  k_fill_f32<<<dim3(1), 64, 0, stream>>>(BNS, 0.f, 64);  // fills BNS and BN2 (contiguous)
  k_post_stats<<<dim3((unsigned)cdivl(N, 8)), 256, 0, stream>>>(G, gat_b, Y, BNS, BN2, (int)N);
  k_bn_norm<<<GRID(n32), TB, 0, stream>>>(Y, BNS, BN2, bn_g, bn_b, (float*)d_out, (int)N);
#undef GRID
}